// GNNBase_78847009620727
// MI455X (gfx1250) — compile-verified
//
#include <hip/hip_runtime.h>
#include <hip/hip_bf16.h>
#include <math.h>

typedef __attribute__((ext_vector_type(2))) float v2f;
typedef __attribute__((ext_vector_type(8))) float v8f;

#define DH 128   // feature dim (= H*HD)
#define NH 4     // heads
#define HD 32    // per-head dim
#define LDS_STRIDE 132  // padded row stride (floats) to avoid LDS bank conflicts

// ---- monotone float <-> u32 mapping so unsigned atomicMax == float max ----
__device__ __forceinline__ unsigned f2u_mono(float f) {
  unsigned u = __float_as_uint(f);
  return (u & 0x80000000u) ? ~u : (u | 0x80000000u);
}
__device__ __forceinline__ float u2f_mono(unsigned u) {
  return (u & 0x80000000u) ? __uint_as_float(u & 0x7FFFFFFFu) : __uint_as_float(~u);
}
#define MONO_NEG_INF 0x007FFFFFu   // f2u_mono(-inf)

// ---------------- GEMM: HX = X * W^T  via V_WMMA_F32_16X16X4_F32 ----------
// One wave -> one 16x16 tile of HX. Block = 256 threads = 8 waves = 16 rows x 128 cols.
// A tile (16x128 f32) is staged into LDS once per block with the gfx1250
// async DMA (GLOBAL_LOAD_ASYNC_TO_LDS_B128, tracked by ASYNCcnt).
__global__ void gat_gemm_wmma(const float* __restrict__ X,
                              const float* __restrict__ W,
                              float* __restrict__ HX, int n_nodes) {
  __shared__ float tileX[16 * LDS_STRIDE];

  const int wave = threadIdx.x >> 5;        // 0..7 -> column tile
  const int lane = threadIdx.x & 31;
  const int half = lane >> 4;               // 0/1 -> K sub-pair
  const int lm   = lane & 15;               // M (for A) / N (for B)
  const int row0 = blockIdx.x * 16;
  const int col0 = wave * 16;
  if (row0 >= n_nodes) return;              // uniform over the block

  // ---- async-DMA stage of the A tile: 16 rows x 512B, 16B per lane-pass ----
  {
    const int t = threadIdx.x;
    const int rr = t >> 4;                       // 0..15: tile row
    const int rclamp = (row0 + rr < n_nodes) ? (row0 + rr) : (n_nodes - 1);
    const float* grow = X + (size_t)rclamp * DH;
    #pragma unroll
    for (int p = 0; p < 2; ++p) {
      const int cc = (p * 16 + (t & 15)) * 4;    // float column of a 16B chunk
      const float* g = grow + cc;
      unsigned lds = (unsigned)(size_t)(&tileX[rr * LDS_STRIDE + cc]);
      asm volatile("global_load_async_to_lds_b128 %0, %1, off"
                   :: "v"(lds), "v"(g) : "memory");
    }
    asm volatile("s_wait_asynccnt 0x0" ::: "memory");
    __syncthreads();
  }

  const float* __restrict__ arow = &tileX[lm * LDS_STRIDE];       // A row (M = lm)
  const float* __restrict__ wrow = W + (size_t)(col0 + lm) * DH;  // B col (N = lm)
  __builtin_prefetch(wrow, 0, 3);

  v8f acc = {};
  #pragma unroll 8
  for (int k = 0; k < DH; k += 4) {
    v2f a, b;
    // A 16x4: lanes 0-15 hold K=k,k+1 ; lanes 16-31 hold K=k+2,k+3 (from LDS)
    a.x = arow[k + 2 * half];
    a.y = arow[k + 2 * half + 1];
    // B 4x16 mirrored layout, W^T[k,n] = W[n,k]
    b.x = wrow[k + 2 * half];
    b.y = wrow[k + 2 * half + 1];
    acc = __builtin_amdgcn_wmma_f32_16x16x4_f32(false, a, false, b,
                                                (short)0, acc, false, false);
  }

  // D 16x16: vgpr i -> row (i + 8*half), col = lm
  if (row0 + 16 <= n_nodes) {               // hot path: no per-row guards
    float* drow = HX + (size_t)(row0 + 8 * half) * DH + col0 + lm;
    #pragma unroll
    for (int i = 0; i < 8; ++i) drow[(size_t)i * DH] = acc[i];
  } else {
    #pragma unroll
    for (int i = 0; i < 8; ++i) {
      int rr = row0 + i + 8 * half;
      if (rr < n_nodes) HX[(size_t)rr * DH + col0 + lm] = acc[i];
    }
  }
}

// ---------------- per-node attention dots: al/ar [N,H] --------------------
__global__ void gat_node_att(const float* __restrict__ HX,
                             const float* __restrict__ attl,
                             const float* __restrict__ attr,
                             float* __restrict__ AL, float* __restrict__ AR,
                             int n_nodes) {
  int idx = blockIdx.x * blockDim.x + threadIdx.x;
  if (idx >= n_nodes * NH) return;
  int n = idx >> 2, h = idx & 3;
  const float* hp = HX + (size_t)n * DH + h * HD;
  const float* lp = attl + h * HD;
  const float* rp = attr + h * HD;
  float sl = 0.f, sr = 0.f;
  #pragma unroll
  for (int d = 0; d < HD; ++d) { float v = hp[d]; sl += v * lp[d]; sr += v * rp[d]; }
  AL[idx] = sl; AR[idx] = sr;
}

// ---------------- init node max / sum / output accumulator ----------------
__global__ void gat_init(unsigned* __restrict__ nmax_u, float* __restrict__ nsum,
                         float* __restrict__ ACC, int n_nodes) {
  int idx = blockIdx.x * blockDim.x + threadIdx.x;
  if (idx < n_nodes * NH) { nmax_u[idx] = MONO_NEG_INF; nsum[idx] = 0.f; }
  if (idx < n_nodes * DH) ACC[idx] = 0.f;
}

// ---------------- edge pass 1: leaky-relu attn + segment max --------------
__global__ void gat_edge_max(const int* __restrict__ ei, int E,
                             const float* __restrict__ AL,
                             const float* __restrict__ AR,
                             float* __restrict__ EATT,
                             unsigned* __restrict__ nmax_u) {
  int idx = blockIdx.x * blockDim.x + threadIdx.x;
  if (idx >= E * NH) return;
  int e = idx >> 2, h = idx & 3;
  int s = ei[e], d = ei[E + e];
  float a = AL[s * NH + h] + AR[d * NH + h];
  a = fmaxf(a, 0.2f * a);                    // leaky_relu(0.2)
  EATT[idx] = a;
  atomicMax(&nmax_u[d * NH + h], f2u_mono(a));
}

// ---------------- edge pass 2: exp + segment sum --------------------------
__global__ void gat_edge_exp(const int* __restrict__ ei, int E,
                             float* __restrict__ EATT,
                             const unsigned* __restrict__ nmax_u,
                             float* __restrict__ nsum) {
  int idx = blockIdx.x * blockDim.x + threadIdx.x;
  if (idx >= E * NH) return;
  int e = idx >> 2, h = idx & 3;
  int d = ei[E + e];
  float m = u2f_mono(nmax_u[d * NH + h]);
  float v = __expf(EATT[idx] - m);
  EATT[idx] = v;
  atomicAdd(&nsum[d * NH + h], v);
}

// ---------------- edge pass 3: normalized message scatter -----------------
// One wave32 per (edge, head): lane = feature within head. Contiguous 128B
// gather of hx[src] and contiguous 128B atomic burst to ACC[dst].
__global__ void gat_scatter(const int* __restrict__ ei, int E,
                            const float* __restrict__ HX,
                            const float* __restrict__ EATT,
                            const float* __restrict__ nsum,
                            float* __restrict__ ACC) {
  int unit = blockIdx.x * (blockDim.x >> 5) + (threadIdx.x >> 5);
  int lane = threadIdx.x & 31;
  if (unit >= E * NH) return;
  int e = unit >> 2, h = unit & 3;
  int s = ei[e], d = ei[E + e];
  float w = EATT[unit] / (nsum[d * NH + h] + 1e-9f);
  float v = w * HX[(size_t)s * DH + h * HD + lane];
  atomicAdd(&ACC[(size_t)d * DH + h * HD + lane], v);
}

// ---------------- layer-0 epilogue: x1 = relu(acc + b0) -------------------
__global__ void gat_bias_relu(float* __restrict__ ACC, const float* __restrict__ b,
                              int n_nodes) {
  int idx = blockIdx.x * blockDim.x + threadIdx.x;
  if (idx >= n_nodes * DH) return;
  float v = ACC[idx] + b[idx & (DH - 1)];
  ACC[idx] = fmaxf(v, 0.f);
}

// ---------------- layer-1 epilogue: head mean + b1 ------------------------
__global__ void gat_head_mean(const float* __restrict__ ACC,
                              const float* __restrict__ b1,
                              float* __restrict__ out, int n_nodes) {
  int idx = blockIdx.x * blockDim.x + threadIdx.x;
  if (idx >= n_nodes * HD) return;
  int n = idx >> 5, d = idx & 31;
  const float* p = ACC + (size_t)n * DH + d;
  out[idx] = 0.25f * (p[0] + p[HD] + p[2 * HD] + p[3 * HD]) + b1[d];
}

extern "C" void kernel_launch(void* const* d_in, const int* in_sizes, int n_in,
                              void* d_out, int out_size, void* d_ws, size_t ws_size,
                              hipStream_t stream) {
  const float* x     = (const float*)d_in[0];
  const int*   ei    = (const int*)d_in[1];
  const float* W0    = (const float*)d_in[2];
  const float* attl0 = (const float*)d_in[3];
  const float* attr0 = (const float*)d_in[4];
  const float* b0    = (const float*)d_in[5];
  const float* W1    = (const float*)d_in[6];
  const float* attl1 = (const float*)d_in[7];
  const float* attr1 = (const float*)d_in[8];
  const float* b1    = (const float*)d_in[9];
  float* out = (float*)d_out;

  const int N = in_sizes[0] / DH;   // 50000
  const int E = in_sizes[1] / 2;    // 800000

  // workspace carve-up (~67 MB)
  float* A  = (float*)d_ws;                 // hx       [N,128]
  float* B  = A + (size_t)N * DH;           // acc / x1 [N,128]
  float* AL = B + (size_t)N * DH;           // [N,4]
  float* AR = AL + (size_t)N * NH;          // [N,4]
  float* NMAXf = AR + (size_t)N * NH;       // [N,4] (u32 view)
  unsigned* NMAX = (unsigned*)NMAXf;
  float* NSUM = NMAXf + (size_t)N * NH;     // [N,4]
  float* EATT = NSUM + (size_t)N * NH;      // [E,4]

  const dim3 blk(256);
  const int g_gemm = (N + 15) / 16;
  const int g_nh   = (N * NH + 255) / 256;
  const int g_nd   = (N * DH + 255) / 256;
  const int g_eh   = (E * NH + 255) / 256;
  const int g_sc   = (E * NH + 7) / 8;      // 8 wave-units per 256-thread block
  const int g_out  = (N * HD + 255) / 256;

  for (int layer = 0; layer < 2; ++layer) {
    const float* W  = layer ? W1 : W0;
    const float* al = layer ? attl1 : attl0;
    const float* ar = layer ? attr1 : attr0;
    const float* xin = layer ? B : x;   // layer-1 input lives in B until GEMM done

    gat_gemm_wmma<<<g_gemm, blk, 0, stream>>>(xin, W, A, N);
    gat_node_att<<<g_nh, blk, 0, stream>>>(A, al, ar, AL, AR, N);
    gat_init<<<g_nd, blk, 0, stream>>>(NMAX, NSUM, B, N);   // B free now (stream-ordered)
    gat_edge_max<<<g_eh, blk, 0, stream>>>(ei, E, AL, AR, EATT, NMAX);
    gat_edge_exp<<<g_eh, blk, 0, stream>>>(ei, E, EATT, NMAX, NSUM);
    gat_scatter<<<g_sc, blk, 0, stream>>>(ei, E, A, EATT, NSUM, B);
    if (layer == 0) {
      gat_bias_relu<<<g_nd, blk, 0, stream>>>(B, b0, N);
    } else {
      gat_head_mean<<<g_out, blk, 0, stream>>>(B, b1, out, N);
    }
  }
}